// Pulse_58755152609819
// MI455X (gfx1250) — compile-verified
//
#include <hip/hip_runtime.h>

// ---------------------------------------------------------------------------
// 12-qubit tensor-product unitary applied to a (4096 x 512) complex64 batch.
// Factor M = kron(A, B), A = kron(U11..U6), B = kron(U5..U0)  (64x64 complex).
// One fused pass per 4-column tile, LDS-resident, fp32 WMMA 16x16x4 matmuls.
// ---------------------------------------------------------------------------

typedef __attribute__((ext_vector_type(2))) float v2f;
typedef __attribute__((ext_vector_type(8))) float v8f;

#define DIM   4096
#define BATCH 512
#define NQ    12
#define CT    4            // batch columns per workgroup
#define NWG   (BATCH/CT)   // 128 workgroups

static __device__ __forceinline__ v8f wmma4(v2f a, v2f b, v8f c) {
  // D = A(16x4) x B(4x16) + C, fp32.  8-arg form: (neg_a, A, neg_b, B, c_mod, C, reuse_a, reuse_b)
  return __builtin_amdgcn_wmma_f32_16x16x4_f32(false, a, false, b, (short)0, c,
                                               false, false);
}

// ---------------------------------------------------------------------------
// Prep kernel: build Ar/Ai/Br/Bi (each 64x64 f32, row-major) into ws.
// ws layout (floats): Ar @0, Ai @4096, Br @8192, Bi @12288   (64 KB total)
// ---------------------------------------------------------------------------
__global__ void build_ops(const float* __restrict__ params, float* __restrict__ ws) {
  __shared__ float g[NQ][2][2][2];  // [qubit][row][col][re/im]
  const int t = threadIdx.x;
  if (t < NQ) {
    const float th = params[t * 3 + 0];
    const float ph = params[t * 3 + 1];
    const float la = params[t * 3 + 2];
    const float c = cosf(th * 0.5f), s = sinf(th * 0.5f);
    const float cl = cosf(la), sl = sinf(la);
    const float cp = cosf(ph), sp = sinf(ph);
    const float cq = cosf(ph + la), sq = sinf(ph + la);
    g[t][0][0][0] = c;        g[t][0][0][1] = 0.0f;
    g[t][0][1][0] = -s * cl;  g[t][0][1][1] = -s * sl;
    g[t][1][0][0] = s * cp;   g[t][1][0][1] = s * sp;
    g[t][1][1][0] = c * cq;   g[t][1][1][1] = c * sq;
  }
  __syncthreads();
  for (int e = t; e < 64 * 64; e += blockDim.x) {
    const int i = e >> 6, j = e & 63;
    float ar = 1.0f, ai = 0.0f, br = 1.0f, bi = 0.0f;
    #pragma unroll
    for (int k = 0; k < 6; ++k) {
      const int ri = (i >> k) & 1, cj = (j >> k) & 1;
      // A factor: qubit 6+k
      float ur = g[6 + k][ri][cj][0], ui = g[6 + k][ri][cj][1];
      float nr = ar * ur - ai * ui, ni = ar * ui + ai * ur;
      ar = nr; ai = ni;
      // B factor: qubit k
      ur = g[k][ri][cj][0]; ui = g[k][ri][cj][1];
      nr = br * ur - bi * ui; ni = br * ui + bi * ur;
      br = nr; bi = ni;
    }
    ws[e] = ar; ws[4096 + e] = ai; ws[8192 + e] = br; ws[12288 + e] = bi;
  }
}

// ---------------------------------------------------------------------------
// Main kernel: one workgroup = 4 batch columns, 256 threads = 8 waves.
// Dynamic LDS: A/B ops (16384 f) + Sr (16384 f) + Si (16384 f) = 192 KB.
// ---------------------------------------------------------------------------
__global__ __launch_bounds__(256)
void kron_apply(const float* __restrict__ xre, const float* __restrict__ xim,
                const float* __restrict__ ws, float* __restrict__ out) {
  extern __shared__ float lds[];
  float* Ar = lds;
  float* Ai = lds + 4096;
  float* Br = lds + 8192;
  float* Bi = lds + 12288;
  float* Sr = lds + 16384;  // holds X tile, then step-A output S
  float* Si = lds + 32768;

  const int t    = threadIdx.x;
  const int lane = t & 31;
  const int wave = t >> 5;           // 0..7
  const int cb   = blockIdx.x * CT;  // first batch column of this tile

  // Stage operator matrices (64 KB) into LDS.
  for (int e = t; e < 4096; e += 256)
    ((float4*)lds)[e] = ((const float4*)ws)[e];
  // Stage x tile: X[(64*i' + j')*CT + c] = x[row, cb+c]; 4 contiguous floats/row.
  for (int r = t; r < DIM; r += 256) {
    *(float4*)(Sr + r * CT) = *(const float4*)(xre + (size_t)r * BATCH + cb);
    *(float4*)(Si + r * CT) = *(const float4*)(xim + (size_t)r * BATCH + cb);
  }
  __syncthreads();

  // Tile assignment: M (64) -> 4 tiles, N (64*CT=256) -> 16 tiles; 8 per wave.
  const int m     = wave & 3;
  const int nbase = (wave >> 2) * 8;          // 0 or 8
  const int rowA  = m * 16 + (lane & 15);     // operator row this lane supplies
  const int khalf = (lane >> 4) << 1;         // K sub-offset {0,2} per lane half
  const int voff  = (lane >> 4) << 3;         // D-layout M offset {0,8}

  // ---------------- Step A: S = A (complex) @ Xhat, K = i' -----------------
  v8f accR[8] = {};
  v8f accI[8] = {};
  for (int k = 0; k < 64; k += 4) {
    const int kb = k + khalf;
    const int ao = rowA * 64 + kb;
    v2f ar; ar.x = Ar[ao]; ar.y = Ar[ao + 1];
    v2f ai; ai.x = Ai[ao]; ai.y = Ai[ao + 1];
    const v2f nai = -ai;
    #pragma unroll
    for (int q = 0; q < 8; ++q) {
      const int col = (nbase + q) * 16 + (lane & 15);        // (j', c) packed
      const int xo  = kb * 256 + (col >> 2) * 4 + (col & 3); // Xhat[K=i'][col]
      v2f xr; xr.x = Sr[xo]; xr.y = Sr[xo + 256];
      v2f xi; xi.x = Si[xo]; xi.y = Si[xo + 256];
      accR[q] = wmma4(ar,  xr, accR[q]);   // Sr += Ar*Xr
      accR[q] = wmma4(nai, xi, accR[q]);   // Sr -= Ai*Xi
      accI[q] = wmma4(ar,  xi, accI[q]);   // Si += Ar*Xi
      accI[q] = wmma4(ai,  xr, accI[q]);   // Si += Ai*Xr
    }
  }
  __syncthreads();  // X fully consumed; safe to overwrite with S

  // Scatter S into Shat[j'][i][c] layout: addr = j'*256 + i*4 + c.
  #pragma unroll
  for (int q = 0; q < 8; ++q) {
    const int col = (nbase + q) * 16 + (lane & 15);
    const int jp = col >> 2, c = col & 3;
    #pragma unroll
    for (int v = 0; v < 8; ++v) {
      const int i  = m * 16 + v + voff;   // D layout: M = v + 8*(lane>=16)
      const int so = jp * 256 + i * 4 + c;
      Sr[so] = accR[q][v];
      Si[so] = accI[q][v];
    }
  }
  __syncthreads();

  // ---------------- Step B: Y = B (complex) @ Shat, K = j' -----------------
  v8f yR[8] = {};
  v8f yI[8] = {};
  for (int k = 0; k < 64; k += 4) {
    const int kb = k + khalf;
    const int bo = rowA * 64 + kb;
    v2f br; br.x = Br[bo]; br.y = Br[bo + 1];
    v2f bi; bi.x = Bi[bo]; bi.y = Bi[bo + 1];
    const v2f nbi = -bi;
    #pragma unroll
    for (int q = 0; q < 8; ++q) {
      const int col = (nbase + q) * 16 + (lane & 15);        // (i, c) packed
      const int so  = kb * 256 + (col >> 2) * 4 + (col & 3); // Shat[K=j'][col]
      v2f sr; sr.x = Sr[so]; sr.y = Sr[so + 256];
      v2f si; si.x = Si[so]; si.y = Si[so + 256];
      yR[q] = wmma4(br,  sr, yR[q]);
      yR[q] = wmma4(nbi, si, yR[q]);
      yI[q] = wmma4(br,  si, yI[q]);
      yI[q] = wmma4(bi,  sr, yI[q]);
    }
  }

  // Store interleaved complex64: out[(r*BATCH + col)*2 + {0,1}], r = i*64 + j.
  #pragma unroll
  for (int q = 0; q < 8; ++q) {
    const int col = (nbase + q) * 16 + (lane & 15);
    const int i = col >> 2, c = col & 3;
    #pragma unroll
    for (int v = 0; v < 8; ++v) {
      const int j = m * 16 + v + voff;
      const size_t r = (size_t)(i * 64 + j);
      float2 val; val.x = yR[q][v]; val.y = yI[q][v];
      *(float2*)(out + (r * BATCH + cb + c) * 2) = val;
    }
  }
}

// ---------------------------------------------------------------------------
extern "C" void kernel_launch(void* const* d_in, const int* in_sizes, int n_in,
                              void* d_out, int out_size, void* d_ws, size_t ws_size,
                              hipStream_t stream) {
  (void)in_sizes; (void)n_in; (void)out_size; (void)ws_size;
  const float* params = (const float*)d_in[0];  // (12, 3) f32
  const float* xre    = (const float*)d_in[1];  // (4096, 512) f32
  const float* xim    = (const float*)d_in[2];  // (4096, 512) f32
  float* out = (float*)d_out;                   // (4096, 512) complex64 interleaved
  float* ws  = (float*)d_ws;                    // needs 64 KB for Ar/Ai/Br/Bi

  build_ops<<<dim3(1), dim3(256), 0, stream>>>(params, ws);

  const int lds_bytes = 49152 * (int)sizeof(float);  // 192 KB dynamic LDS
  hipFuncSetAttribute(reinterpret_cast<const void*>(kron_apply),
                      hipFuncAttributeMaxDynamicSharedMemorySize, lds_bytes);
  kron_apply<<<dim3(NWG), dim3(256), lds_bytes, stream>>>(xre, xim, ws, out);
}